// GraphSAGE_2199023255751
// MI455X (gfx1250) — compile-verified
//
#include <hip/hip_runtime.h>

typedef float v2f __attribute__((ext_vector_type(2)));
typedef float v8f __attribute__((ext_vector_type(8)));

// ---------------------------------------------------------------------------
// Zero fill (grid-stride)
// ---------------------------------------------------------------------------
__global__ void fill_zero_kernel(float* __restrict__ p, long long n) {
    long long i = (long long)blockIdx.x * blockDim.x + threadIdx.x;
    long long stride = (long long)gridDim.x * blockDim.x;
    for (; i < n; i += stride) p[i] = 0.0f;
}

// ---------------------------------------------------------------------------
// Degree: deg[dst[e]] += 1
// ---------------------------------------------------------------------------
__global__ void degree_kernel(const long long* __restrict__ dst,
                              float* __restrict__ deg, int E) {
    int e = blockIdx.x * blockDim.x + threadIdx.x;
    if (e < E) atomicAdd(&deg[(int)dst[e]], 1.0f);
}

// ---------------------------------------------------------------------------
// Scatter-add: agg[dst[e], f] += feat[src[e], f]   (thread per (e,f))
// ---------------------------------------------------------------------------
template <int D>
__global__ void scatter_add_kernel(const long long* __restrict__ src,
                                   const long long* __restrict__ dst,
                                   const float* __restrict__ feat,
                                   float* __restrict__ agg, long long total) {
    long long idx = (long long)blockIdx.x * blockDim.x + threadIdx.x;
    if (idx >= total) return;
    int e = (int)(idx / D);
    int f = (int)(idx - (long long)e * D);
    int s = (int)src[e];
    int d = (int)dst[e];
    atomicAdd(&agg[(long long)d * D + f], feat[(long long)s * D + f]);
}

// ---------------------------------------------------------------------------
// Fused SAGE layer GEMM:
//   out[i,:] = act( (agg[i,:]/max(deg[i],1)) @ Wl + x[i,:] @ Wr + b )
// 16 node-rows per block; one wave per 16-column output tile.
// Uses V_WMMA_F32_16X16X4_F32 (fp32 in/out, wave32 matrix pipe).
// ---------------------------------------------------------------------------
template <int K, int NCOL>
__global__ __launch_bounds__(32 * (NCOL / 16))
void sage_gemm_kernel(const float* __restrict__ agg,
                      const float* __restrict__ deg,
                      const float* __restrict__ xin,
                      const float* __restrict__ Wl,
                      const float* __restrict__ Wr,
                      const float* __restrict__ bias,
                      float* __restrict__ out,
                      int n_nodes, int do_relu) {
    constexpr int NW   = NCOL / 16;   // waves per block
    constexpr int ASTR = 2 * K + 1;   // padded LDS row stride (bank-conflict free)
    __shared__ float A[16 * ASTR];

    const int tid  = threadIdx.x;
    const int row0 = blockIdx.x * 16;

    // Stage A block into LDS: [agg * invdeg | x], 16 rows x 2K cols.
    for (int idx = tid; idx < 16 * K; idx += 32 * NW) {
        int r = idx / K;
        int k = idx - r * K;
        int node = row0 + r;
        float va = 0.0f, vx = 0.0f;
        if (node < n_nodes) {
            float dg  = deg[node];
            float inv = 1.0f / fmaxf(dg, 1.0f);
            va = agg[(long long)node * K + k] * inv;
            vx = xin[(long long)node * K + k];
        }
        A[r * ASTR + k]     = va;
        A[r * ASTR + K + k] = vx;
    }
    __syncthreads();

    const int wave = tid >> 5;
    const int lane = tid & 31;
    const int m    = lane & 15;     // M for A-frag, N for B/C/D-frag
    const int half = lane >> 4;     // selects K-pair (A/B) and M-half (C/D)
    const int kb   = half * 2;
    const int col  = wave * 16 + m;

    // C init = bias broadcast (every accumulator VGPR holds same column).
    v8f c;
    {
        float bv = bias[col];
#pragma unroll
        for (int v = 0; v < 8; ++v) c[v] = bv;
    }

    // Pass 1: (agg/deg) @ Wl
#pragma unroll
    for (int k0 = 0; k0 < K; k0 += 4) {
        v2f a, b;
        a.x = A[m * ASTR + k0 + kb];
        a.y = A[m * ASTR + k0 + kb + 1];
        b.x = Wl[(k0 + kb) * NCOL + col];
        b.y = Wl[(k0 + kb + 1) * NCOL + col];
        c = __builtin_amdgcn_wmma_f32_16x16x4_f32(false, a, false, b,
                                                  (short)0, c, false, false);
    }
    // Pass 2: x @ Wr
#pragma unroll
    for (int k0 = 0; k0 < K; k0 += 4) {
        v2f a, b;
        a.x = A[m * ASTR + K + k0 + kb];
        a.y = A[m * ASTR + K + k0 + kb + 1];
        b.x = Wr[(k0 + kb) * NCOL + col];
        b.y = Wr[(k0 + kb + 1) * NCOL + col];
        c = __builtin_amdgcn_wmma_f32_16x16x4_f32(false, a, false, b,
                                                  (short)0, c, false, false);
    }

    // D layout: VGPR v, lanes 0-15 -> M=v, lanes 16-31 -> M=v+8.
#pragma unroll
    for (int v = 0; v < 8; ++v) {
        int node = row0 + v + 8 * half;
        if (node < n_nodes) {
            float val = c[v];
            if (do_relu) val = fmaxf(val, 0.0f);
            out[(long long)node * NCOL + col] = val;
        }
    }
}

// ---------------------------------------------------------------------------
// Host-side launch
// ---------------------------------------------------------------------------
extern "C" void kernel_launch(void* const* d_in, const int* in_sizes, int n_in,
                              void* d_out, int out_size, void* d_ws, size_t ws_size,
                              hipStream_t stream) {
    (void)n_in; (void)out_size; (void)ws_size;

    const float*     x   = (const float*)d_in[0];
    const long long* ei  = (const long long*)d_in[1];  // int64 edge_index [2,E]
    const float*     Wl1 = (const float*)d_in[2];      // [64,128]
    const float*     Wr1 = (const float*)d_in[3];      // [64,128]
    const float*     b1  = (const float*)d_in[4];      // [128]
    const float*     Wl2 = (const float*)d_in[5];      // [128,64]
    const float*     Wr2 = (const float*)d_in[6];      // [128,64]
    const float*     b2  = (const float*)d_in[7];      // [64]
    float*           out = (float*)d_out;

    const int N = in_sizes[0] / 64;
    const int E = in_sizes[1] / 2;
    const long long* src = ei;
    const long long* dst = ei + E;

    // Workspace layout: h [N*128] | agg [N*128] | deg [N]
    float* h   = (float*)d_ws;
    float* agg = h + (size_t)N * 128;
    float* deg = agg + (size_t)N * 128;

    const int ZB = 4096;   // grid-stride zero blocks

    // ---- Layer 1 ----
    fill_zero_kernel<<<ZB, 256, 0, stream>>>(agg, (long long)N * 128 + N);
    degree_kernel<<<(E + 255) / 256, 256, 0, stream>>>(dst, deg, E);
    {
        long long total = (long long)E * 64;
        int blocks = (int)((total + 255) / 256);
        scatter_add_kernel<64><<<blocks, 256, 0, stream>>>(src, dst, x, agg, total);
    }
    sage_gemm_kernel<64, 128><<<(N + 15) / 16, 256, 0, stream>>>(
        agg, deg, x, Wl1, Wr1, b1, h, N, /*relu=*/1);

    // ---- Layer 2 ----
    fill_zero_kernel<<<ZB, 256, 0, stream>>>(agg, (long long)N * 128);
    {
        long long total = (long long)E * 128;
        int blocks = (int)((total + 255) / 256);
        scatter_add_kernel<128><<<blocks, 256, 0, stream>>>(src, dst, h, agg, total);
    }
    sage_gemm_kernel<128, 64><<<(N + 15) / 16, 128, 0, stream>>>(
        agg, deg, h, Wl2, Wr2, b2, out, N, /*relu=*/0);
}